// SignatureScoringLoss_60284160967036
// MI455X (gfx1250) — compile-verified
//
#include <hip/hip_runtime.h>
#include <hip/hip_bf16.h>

typedef float v2f __attribute__((ext_vector_type(2)));
typedef float v8f __attribute__((ext_vector_type(8)));

#define SEQ_L   65          // time points per sequence
#define NSEQ    64          // SEQ_L - 1 (base increment grid)
#define S_PDE   128         // dyadic-refined grid (F=2)
#define XSTR    80          // padded LDS stride for staged sequences (5 tiles of 16)
#define GSTR    81          // padded Gram stride
#define GROWS   80
#define M_POP   16
#define B_SZ    4
#define NPAIR_XX (B_SZ * M_POP * M_POP)       // 1024
#define NPAIR    (NPAIR_XX + B_SZ * M_POP)    // 1088

// One wave32 per (X,Y) sequence pair: WMMA Gram -> exp -> increments -> wavefront PDE.
__global__ __launch_bounds__(32)
void sigkernel_pair_kernel(const float* __restrict__ gen,
                           const float* __restrict__ realp,
                           float* __restrict__ pairRes) {
    __shared__ float Xs[8 * XSTR];          // X, d-major, zero padded cols 65..79
    __shared__ float Ys[8 * XSTR];
    __shared__ float sx[XSTR];              // ||x_s||^2
    __shared__ float sy[XSTR];
    __shared__ float G[GROWS * GSTR];       // exp'd RBF Gram (only [0..64]^2 meaningful)
    __shared__ float incB[NSEQ * NSEQ];     // base double-increments, pre-scaled by 1/4

    const int lane = threadIdx.x;
    const int p    = blockIdx.x;

    // ---- pair -> sequence pointers (layout: (B, m, dim, L), time fastest) ----
    const float* Xp;
    const float* Yp;
    if (p < NPAIR_XX) {
        int b = p >> 8;
        int i = (p >> 4) & 15;
        int j = p & 15;
        Xp = gen + (size_t)(b * M_POP + i) * 8 * SEQ_L;
        Yp = gen + (size_t)(b * M_POP + j) * 8 * SEQ_L;
    } else {
        int q = p - NPAIR_XX;
        int b = q >> 4;
        int i = q & 15;
        Xp = gen + (size_t)(b * M_POP + i) * 8 * SEQ_L;
        Yp = realp + (size_t)b * 8 * SEQ_L;
    }

    // ---- stage sequences into padded LDS ----
    for (int idx = lane; idx < 8 * XSTR; idx += 32) { Xs[idx] = 0.0f; Ys[idx] = 0.0f; }
    __syncthreads();
    for (int idx = lane; idx < 8 * SEQ_L; idx += 32) {
        int d = idx / SEQ_L, s = idx % SEQ_L;
        Xs[d * XSTR + s] = Xp[idx];
        Ys[d * XSTR + s] = Yp[idx];
    }
    __syncthreads();
    for (int s = lane; s < XSTR; s += 32) {
        float ax = 0.0f, ay = 0.0f;
#pragma unroll
        for (int d = 0; d < 8; ++d) {
            float xv = Xs[d * XSTR + s]; ax += xv * xv;
            float yv = Ys[d * XSTR + s]; ay += yv * yv;
        }
        sx[s] = ax; sy[s] = ay;
    }
    __syncthreads();

    // ---- RBF Gram via V_WMMA_F32_16X16X4_F32: inner = X^T (65x8) * Y (8x65) ----
    // A 16x4 f32 layout: lanes 0-15 hold M=lane, K={0,1}; lanes 16-31 hold M=lane-16, K={2,3}
    const int mrow  = lane & 15;
    const int khalf = (lane >> 4) << 1;      // 0 or 2
    const int rowhi = (lane >> 4) << 3;      // D layout: +8 rows for upper half-wave
    for (int si = 0; si < 5; ++si) {
        for (int tj = 0; tj < 5; ++tj) {
            const int srow = si * 16 + mrow;
            const int tcol = tj * 16 + mrow;
            v8f acc = {0.f, 0.f, 0.f, 0.f, 0.f, 0.f, 0.f, 0.f};
#pragma unroll
            for (int kb = 0; kb < 8; kb += 4) {
                v2f av, bv;
                av.x = Xs[(kb + khalf + 0) * XSTR + srow];
                av.y = Xs[(kb + khalf + 1) * XSTR + srow];
                bv.x = Ys[(kb + khalf + 0) * XSTR + tcol];
                bv.y = Ys[(kb + khalf + 1) * XSTR + tcol];
                acc = __builtin_amdgcn_wmma_f32_16x16x4_f32(
                          false, av, false, bv, (short)0, acc, false, false);
            }
#pragma unroll
            for (int r = 0; r < 8; ++r) {
                int gs = si * 16 + r + rowhi;
                float d2 = sx[gs] + sy[tcol] - 2.0f * acc[r];
                G[gs * GSTR + tcol] = __expf(-d2);   // SIGMA = 1
            }
        }
    }
    __syncthreads();

    // ---- base double increments, pre-divided by F*F = 4 ----
    for (int idx = lane; idx < NSEQ * NSEQ; idx += 32) {
        int s = idx >> 6, t = idx & 63;
        float v = G[(s + 1) * GSTR + (t + 1)] + G[s * GSTR + t]
                - G[(s + 1) * GSTR + t]       - G[s * GSTR + (t + 1)];
        incB[idx] = 0.25f * v;
    }
    __syncthreads();

    // ---- Goursat PDE, anti-diagonal wavefront inside one wave32 ----
    // Lane L owns columns 4L+1..4L+4 of the (S+1)x(S+1) grid; at step T it
    // processes row r = T - L (computing K[r+1][*]); the left boundary value
    // flows from lane L-1 via shuffle each step. 159 steps total.
    float kp0 = 1.0f, kp1 = 1.0f, kp2 = 1.0f, kp3 = 1.0f;  // K[r][own cols]
    float inL_old = 1.0f, inL_new = 1.0f;                  // K[r][4L], K[r+1][4L]
    for (int T = 0; T < S_PDE + 31; ++T) {
        int r = T - lane;
        bool active = (r >= 0) && (r < S_PDE);
        int rb = (r < 0 ? 0 : (r >= S_PDE ? S_PDE - 1 : r)) >> 1;
        float m0 = incB[rb * NSEQ + 2 * lane];
        float m1 = incB[rb * NSEQ + 2 * lane + 1];
        if (!active) { m0 = 0.0f; m1 = 0.0f; }
        float a0 = 1.0f + 0.5f * m0 + m0 * m0 * (1.0f / 12.0f);
        float b0 = 1.0f - m0 * m0 * (1.0f / 12.0f);
        float a1 = 1.0f + 0.5f * m1 + m1 * m1 * (1.0f / 12.0f);
        float b1 = 1.0f - m1 * m1 * (1.0f / 12.0f);
        float c1 = (inL_new + kp0) * a0 - inL_old * b0;    // K[r+1][4L+1]
        float c2 = (c1 + kp1) * a0 - kp0 * b0;             // K[r+1][4L+2]
        float c3 = (c2 + kp2) * a1 - kp1 * b1;             // K[r+1][4L+3]
        float c4 = (c3 + kp3) * a1 - kp2 * b1;             // K[r+1][4L+4]
        float recv = __shfl_up(c4, 1, 32);                 // K[next r+1][4L] from left lane
        if (lane == 0) recv = 1.0f;                        // K[*][0] = 1
        bool nextActive = ((T + 1 - lane) >= 0) && ((T + 1 - lane) < S_PDE);
        if (active) {
            kp0 = c1; kp1 = c2; kp2 = c3; kp3 = c4;
            inL_old = inL_new;
            inL_new = recv;
        } else if (nextActive) {                           // activating next step (row 0)
            inL_old = 1.0f;
            inL_new = recv;
        }
    }
    if (lane == 31) pairRes[p] = kp3;                      // K[128][128]
}

// Weighted reduction: mean_b( (LAMBDA/2) * (sum-trace)/(m(m-1)) - mean_i K_XY )
__global__ __launch_bounds__(256)
void sigkernel_reduce_kernel(const float* __restrict__ pairRes,
                             float* __restrict__ out) {
    __shared__ float red[256];
    int t = threadIdx.x;
    float acc = 0.0f;
    for (int p = t; p < NPAIR; p += 256) {
        float v = pairRes[p];
        float w;
        if (p < NPAIR_XX) {
            int i = (p >> 4) & 15, j = p & 15;
            // LAMBDA/2 / (m*(m-1)) / B ; diagonal excluded (trace subtraction)
            w = (i == j) ? 0.0f : (0.5f / (16.0f * 15.0f * 4.0f));
        } else {
            // -1 / m / B
            w = -1.0f / (16.0f * 4.0f);
        }
        acc += w * v;
    }
    red[t] = acc;
    __syncthreads();
    for (int off = 128; off > 0; off >>= 1) {
        if (t < off) red[t] += red[t + off];
        __syncthreads();
    }
    if (t == 0) out[0] = red[0];
}

extern "C" void kernel_launch(void* const* d_in, const int* in_sizes, int n_in,
                              void* d_out, int out_size, void* d_ws, size_t ws_size,
                              hipStream_t stream) {
    const float* gen   = (const float*)d_in[0];   // (4, 16, 8, 65) f32
    const float* realp = (const float*)d_in[1];   // (4, 8, 65) f32
    float* pairRes = (float*)d_ws;                // 1088 floats of scratch
    float* out     = (float*)d_out;               // scalar loss

    sigkernel_pair_kernel<<<NPAIR, 32, 0, stream>>>(gen, realp, pairRes);
    sigkernel_reduce_kernel<<<1, 256, 0, stream>>>(pairRes, out);
}